// PaeGmm_29180007809055
// MI455X (gfx1250) — compile-verified
//
#include <hip/hip_runtime.h>
#include <math.h>

// ---------------------------------------------------------------------------
// Problem constants (from reference): Ur [16384,64], Uc [8192,64], K = 64,
// nclu_row = nclu_col = 64.  loss = log(1 + |1 - mi_red/mi_org|).
//
// mi_org over the full 16384x8192 table is computed fused with the WMMA GEMM
// (T never materialized), using the entropy split
//   mi_org = sum t*log2(t+eps) - sum_i px*log2(px) - sum_j py*log2(py)
// which is valid to f32 precision because eps=1e-15 << px*py ~ 7.5e-9.
// ---------------------------------------------------------------------------
#define N_INS 16384
#define N_ATT 8192
#define KDIM  64
#define EPSF  1e-15f

typedef __attribute__((ext_vector_type(2))) float v2f;
typedef __attribute__((ext_vector_type(8))) float v8f;

// ---------------------------------------------------------------------------
// Workspace layout (floats)
// ---------------------------------------------------------------------------
#define WS_CSUR   0            // [64]    column sums of Ur
#define WS_CSUC   64           // [64]    column sums of Uc
#define WS_ASEG   128          // [64*64] segment-summed Ur rows
#define WS_BSEG   4224         // [64*64] segment-summed Uc rows
#define WS_S      8320         // [1]     S = sum(T) (unnormalized)
#define WS_PX     8448         // [16384] unnormalized row marginals
#define WS_PY     24832        // [8192]  unnormalized col marginals
#define WS_PARTS  33024        // [16384] per-block MI partials

// ---------------------------------------------------------------------------
// Zero the accumulator region [0, 8448)
// ---------------------------------------------------------------------------
__global__ __launch_bounds__(256) void zero_kernel(float* __restrict__ p, int n) {
    int i = blockIdx.x * 256 + threadIdx.x;
    if (i < n) p[i] = 0.0f;
}

// ---------------------------------------------------------------------------
// Per-row: argmax (first max), column sums, segment sums. LDS-staged atomics.
// ---------------------------------------------------------------------------
__global__ __launch_bounds__(256) void row_reduce_kernel(
    const float* __restrict__ U, int nrows,
    float* __restrict__ colsum, float* __restrict__ seg)
{
    __shared__ float lcs[KDIM];
    __shared__ float lseg[KDIM * KDIM];
    for (int t = threadIdx.x; t < KDIM; t += 256)        lcs[t]  = 0.0f;
    for (int t = threadIdx.x; t < KDIM * KDIM; t += 256) lseg[t] = 0.0f;
    __syncthreads();

    int i = blockIdx.x * 256 + threadIdx.x;
    if (i < nrows) {
        const float* row = U + (size_t)i * KDIM;
        float best = row[0];
        int   bi   = 0;
        #pragma unroll
        for (int k = 1; k < KDIM; ++k) {
            float v = row[k];
            if (v > best) { best = v; bi = k; }   // first-occurrence argmax
        }
        #pragma unroll
        for (int k = 0; k < KDIM; ++k) {
            float v = row[k];
            atomicAdd(&lcs[k], v);
            atomicAdd(&lseg[bi * KDIM + k], v);
        }
    }
    __syncthreads();
    for (int t = threadIdx.x; t < KDIM; t += 256) atomicAdd(&colsum[t], lcs[t]);
    for (int t = threadIdx.x; t < KDIM * KDIM; t += 256) {
        float v = lseg[t];
        if (v != 0.0f) atomicAdd(&seg[t], v);
    }
}

// ---------------------------------------------------------------------------
// S = csUr . csUc ; Px[i] = Ur[i].csUc ; Py[j] = Uc[j].csUr  (unnormalized)
// ---------------------------------------------------------------------------
__global__ __launch_bounds__(256) void marginals_kernel(
    const float* __restrict__ Ur, const float* __restrict__ Uc,
    const float* __restrict__ csUr, const float* __restrict__ csUc,
    float* __restrict__ Px, float* __restrict__ Py, float* __restrict__ Sout)
{
    int gid = blockIdx.x * 256 + threadIdx.x;
    if (gid == 0) {
        float s = 0.0f;
        #pragma unroll
        for (int k = 0; k < KDIM; ++k) s += csUr[k] * csUc[k];
        Sout[0] = s;
    }
    if (gid < N_INS) {
        const float* row = Ur + (size_t)gid * KDIM;
        float s = 0.0f;
        #pragma unroll
        for (int k = 0; k < KDIM; ++k) s += row[k] * csUc[k];
        Px[gid] = s;
    } else if (gid < N_INS + N_ATT) {
        const float* row = Uc + (size_t)(gid - N_INS) * KDIM;
        float s = 0.0f;
        #pragma unroll
        for (int k = 0; k < KDIM; ++k) s += row[k] * csUr[k];
        Py[gid - N_INS] = s;
    }
}

// ---------------------------------------------------------------------------
// Main fused kernel: each wave computes a 16(M) x 64(N) strip of T via
// V_WMMA_F32_16X16X4_F32 (4 accumulators share each A fragment), then the
// epilogue folds  t*log2(t+eps)  straight out of the accumulator VGPRs.
// Marginal-entropy terms are handled separately (see finalize_kernel).
//
// A frag (16x4 f32): lane L holds A[M=L%16][K = kh+{0,1}], kh = 2*(L>=16).
// B frag (4x16 f32): lane L holds B[K = kh+{0,1}][N=L%16] — since B = Uc^T,
// lane L loads Uc[n0+L%16][...]: identical pattern to the A load.
// D (16x16 f32, 8 VGPRs): elem r of lane L -> M = r + 8*(L>=16), N = L%16.
// ---------------------------------------------------------------------------
__global__ __launch_bounds__(256) void mi_gemm_kernel(
    const float* __restrict__ Ur, const float* __restrict__ Uc,
    const float* __restrict__ Sptr, float* __restrict__ partials)
{
    const int lane  = threadIdx.x & 31;
    const int wave  = threadIdx.x >> 5;
    const int strip = blockIdx.x * 8 + wave;     // 131072 strips of 16x64
    const int strips_n = N_ATT / 64;             // 128
    const int m0 = (strip / strips_n) * 16;
    const int n0 = (strip % strips_n) * 64;

    const int r  = lane & 15;                    // row-in-tile for frag loads
    const int kh = (lane >> 4) << 1;             // 0 or 2

    const float invS = 1.0f / Sptr[0];

    const float* __restrict__ arow  = Ur + (size_t)(m0 + r) * KDIM + kh;
    const float* __restrict__ brow0 = Uc + (size_t)(n0 + r) * KDIM + kh;
    const float* __restrict__ brow1 = brow0 + 16 * KDIM;
    const float* __restrict__ brow2 = brow0 + 32 * KDIM;
    const float* __restrict__ brow3 = brow0 + 48 * KDIM;

    v8f acc0 = {0.0f, 0.0f, 0.0f, 0.0f, 0.0f, 0.0f, 0.0f, 0.0f};
    v8f acc1 = acc0, acc2 = acc0, acc3 = acc0;

    #pragma unroll
    for (int ks = 0; ks < KDIM / 4; ++ks) {      // 16 K-steps, K=4 each
        v2f a, b0, b1, b2, b3;
        a.x  = arow [ks * 4 + 0];  a.y  = arow [ks * 4 + 1];
        b0.x = brow0[ks * 4 + 0];  b0.y = brow0[ks * 4 + 1];
        b1.x = brow1[ks * 4 + 0];  b1.y = brow1[ks * 4 + 1];
        b2.x = brow2[ks * 4 + 0];  b2.y = brow2[ks * 4 + 1];
        b3.x = brow3[ks * 4 + 0];  b3.y = brow3[ks * 4 + 1];
        acc0 = __builtin_amdgcn_wmma_f32_16x16x4_f32(false, a, false, b0,
                                                     (short)0, acc0, false, false);
        acc1 = __builtin_amdgcn_wmma_f32_16x16x4_f32(false, a, false, b1,
                                                     (short)0, acc1, false, false);
        acc2 = __builtin_amdgcn_wmma_f32_16x16x4_f32(false, a, false, b2,
                                                     (short)0, acc2, false, false);
        acc3 = __builtin_amdgcn_wmma_f32_16x16x4_f32(false, a, false, b3,
                                                     (short)0, acc3, false, false);
    }

    // Fused epilogue: one v_log_f32 per output element, no marginal loads.
    float local = 0.0f;
    #pragma unroll
    for (int e = 0; e < 8; ++e) {
        float t0 = acc0[e] * invS;
        float t1 = acc1[e] * invS;
        float t2 = acc2[e] * invS;
        float t3 = acc3[e] * invS;
        local += t0 * __log2f(t0 + EPSF);
        local += t1 * __log2f(t1 + EPSF);
        local += t2 * __log2f(t2 + EPSF);
        local += t3 * __log2f(t3 + EPSF);
    }

    // wave32 reduce, then cross-wave reduce in LDS.
    #pragma unroll
    for (int off = 16; off > 0; off >>= 1)
        local += __shfl_down(local, off, 32);
    __shared__ float wsum[8];
    if (lane == 0) wsum[wave] = local;
    __syncthreads();
    if (threadIdx.x == 0) {
        float s = 0.0f;
        #pragma unroll
        for (int w = 0; w < 8; ++w) s += wsum[w];
        partials[blockIdx.x] = s;
    }
}

// ---------------------------------------------------------------------------
// Finalize: mi_org = sum(partials) - Hx - Hy ; T_rr = Aseg*Bseg^T/S ; mi_red ;
// loss.  Single block, deterministic tree reductions.
// ---------------------------------------------------------------------------
__global__ __launch_bounds__(256) void finalize_kernel(
    const float* __restrict__ partials, int nparts,
    const float* __restrict__ Px, const float* __restrict__ Py,
    const float* __restrict__ Aseg, const float* __restrict__ Bseg,
    const float* __restrict__ Sptr, float* __restrict__ out)
{
    __shared__ float red[256];
    __shared__ float C[KDIM * KDIM];
    __shared__ float Pxr[KDIM];
    __shared__ float Pyr[KDIM];
    __shared__ float s_mi_org;

    const float invS = 1.0f / Sptr[0];

    // ---- mi_org: joint-term partials minus marginal entropies
    float s = 0.0f;
    for (int i = threadIdx.x; i < nparts; i += 256) s += partials[i];
    for (int i = threadIdx.x; i < N_INS; i += 256) {
        float px = Px[i] * invS;
        s -= px * __log2f(px);
    }
    for (int j = threadIdx.x; j < N_ATT; j += 256) {
        float py = Py[j] * invS;
        s -= py * __log2f(py);
    }
    red[threadIdx.x] = s;
    __syncthreads();
    for (int off = 128; off > 0; off >>= 1) {
        if (threadIdx.x < off) red[threadIdx.x] += red[threadIdx.x + off];
        __syncthreads();
    }
    if (threadIdx.x == 0) s_mi_org = red[0];
    __syncthreads();

    // ---- reduced joint table C = Aseg x Bseg^T / S   [64 x 64]
    for (int e = threadIdx.x; e < KDIM * KDIM; e += 256) {
        int a = e >> 6, b = e & 63;
        float d = 0.0f;
        #pragma unroll
        for (int k = 0; k < KDIM; ++k) d += Aseg[a * KDIM + k] * Bseg[b * KDIM + k];
        C[e] = d * invS;
    }
    __syncthreads();

    // ---- marginals of C
    if (threadIdx.x < KDIM) {
        float rs = 0.0f, cs = 0.0f;
        for (int b = 0; b < KDIM; ++b) {
            rs += C[threadIdx.x * KDIM + b];
            cs += C[b * KDIM + threadIdx.x];
        }
        Pxr[threadIdx.x] = rs;
        Pyr[threadIdx.x] = cs;
    }
    __syncthreads();

    // ---- mi_red (exact per-element form; table is tiny)
    float mr = 0.0f;
    for (int e = threadIdx.x; e < KDIM * KDIM; e += 256) {
        int a = e >> 6, b = e & 63;
        float t = C[e];
        mr += t * (__log2f(t + EPSF) - __log2f(Pxr[a] * Pyr[b] + EPSF));
    }
    red[threadIdx.x] = mr;
    __syncthreads();
    for (int off = 128; off > 0; off >>= 1) {
        if (threadIdx.x < off) red[threadIdx.x] += red[threadIdx.x + off];
        __syncthreads();
    }
    if (threadIdx.x == 0) {
        float mi_red = red[0];
        float mi_org = s_mi_org;
        out[0] = logf(1.0f + fabsf(1.0f - mi_red / mi_org));
    }
}

// ---------------------------------------------------------------------------
// Host-side launcher
// ---------------------------------------------------------------------------
extern "C" void kernel_launch(void* const* d_in, const int* in_sizes, int n_in,
                              void* d_out, int out_size, void* d_ws, size_t ws_size,
                              hipStream_t stream) {
    const float* Ur = (const float*)d_in[0];   // [16384, 64]
    const float* Uc = (const float*)d_in[1];   // [8192, 64]
    // d_in[2], d_in[3]: nclu_row/nclu_col == 64 (hardcoded as KDIM)

    float* ws    = (float*)d_ws;
    float* csUr  = ws + WS_CSUR;
    float* csUc  = ws + WS_CSUC;
    float* Aseg  = ws + WS_ASEG;
    float* Bseg  = ws + WS_BSEG;
    float* Sv    = ws + WS_S;
    float* Px    = ws + WS_PX;
    float* Py    = ws + WS_PY;
    float* parts = ws + WS_PARTS;
    float* outp  = (float*)d_out;

    // 1) zero accumulators (csUr, csUc, Aseg, Bseg, S)
    zero_kernel<<<(8448 + 255) / 256, 256, 0, stream>>>(ws, 8448);

    // 2) per-row reductions: argmax, column sums, segment sums
    row_reduce_kernel<<<N_INS / 256, 256, 0, stream>>>(Ur, N_INS, csUr, Aseg);
    row_reduce_kernel<<<N_ATT / 256, 256, 0, stream>>>(Uc, N_ATT, csUc, Bseg);

    // 3) S, Px, Py
    marginals_kernel<<<(N_INS + N_ATT) / 256, 256, 0, stream>>>(
        Ur, Uc, csUr, csUc, Px, Py, Sv);

    // 4) fused WMMA GEMM + joint-entropy reduction (never materializes T)
    const int nstrips = (N_INS / 16) * (N_ATT / 64);  // 131072 strips
    const int nblocks = nstrips / 8;                  // 16384, 8 waves/block
    mi_gemm_kernel<<<nblocks, 256, 0, stream>>>(Ur, Uc, Sv, parts);

    // 5) finalize: mi_org (+marginal entropies), reduced-table MI, loss
    finalize_kernel<<<1, 256, 0, stream>>>(parts, nblocks, Px, Py,
                                           Aseg, Bseg, Sv, outp);
}